// PNATower_13365938226035
// MI455X (gfx1250) — compile-verified
//
#include <hip/hip_runtime.h>
#include <hip/hip_bf16.h>
#include <math.h>

// Problem constants (from the reference)
#define N_NODES 50000
#define N_EDGES 800000
#define F_IN    64
#define F_ED    16
#define F_OUT   64
#define K_PRE   144          // 2*F_IN + F_ED
#define KG_PRE  36           // K_PRE / 4
#define KG_POST 208          // 832 / 4
#define AVG_D_LOG 2.833f
#define PNA_EPS   1e-5f

// LDS row pitches (even -> 8B-aligned float2 ds loads; odd mod 64 pattern avoids bank conflicts)
#define ZPITCH 146
#define HPITCH 66
#define APITCH 258

typedef __attribute__((ext_vector_type(2))) float v2f;
typedef __attribute__((ext_vector_type(8))) float v8f;

static __device__ __forceinline__ v8f wmma_f32(v2f a, v2f b, v8f c) {
  // D = A(16x4,f32) * B(4x16,f32) + C(16x16,f32); fp32 WMMA keeps reference precision.
  return __builtin_amdgcn_wmma_f32_16x16x4_f32(false, a, false, b, (short)0, c,
                                               false, false);
}

// ---------------------------------------------------------------------------
// prep: zero accumulators, init max/min, swizzle W_pre / W_post into the
// B-operand layout of v_wmma_f32_16x16x4_f32:
//   sw[kg*256 + n*4 + pair*2 + j] = W[(4*kg + 2*pair + j)*64 + n]
// so lane L fetches its float2 {W[k0+off], W[k0+off+1]} (off = 0 or 2) with a
// single 8-byte load.
// ---------------------------------------------------------------------------
__global__ void pna_prep_kernel(const float* __restrict__ Wpre,
                                const float* __restrict__ Wpost,
                                float* __restrict__ sum, float* __restrict__ sumsq,
                                float* __restrict__ mx, float* __restrict__ mn,
                                float* __restrict__ deg,
                                float* __restrict__ swpre, float* __restrict__ swpost) {
  const long long stride = (long long)gridDim.x * blockDim.x;
  const long long t0 = (long long)blockIdx.x * blockDim.x + threadIdx.x;

  for (long long i = t0; i < (long long)N_NODES * F_IN; i += stride) {
    sum[i]   = 0.0f;
    sumsq[i] = 0.0f;
    mx[i]    = -3.402823466e38f;
    mn[i]    =  3.402823466e38f;
  }
  for (long long i = t0; i < N_NODES; i += stride) deg[i] = 0.0f;

  for (long long i = t0; i < (long long)KG_PRE * 256; i += stride) {
    int kg = (int)(i >> 8), rem = (int)(i & 255);
    int n = rem >> 2, q = rem & 3;
    int k = 4 * kg + 2 * (q >> 1) + (q & 1);
    swpre[i] = Wpre[k * F_IN + n];
  }
  for (long long i = t0; i < (long long)KG_POST * 256; i += stride) {
    int kg = (int)(i >> 8), rem = (int)(i & 255);
    int n = rem >> 2, q = rem & 3;
    int k = 4 * kg + 2 * (q >> 1) + (q & 1);
    swpost[i] = Wpost[k * F_OUT + n];
  }
}

// ---------------------------------------------------------------------------
// edge kernel: 4 waves/block, 16 edges per wave.
//   msg[16,64] = z[16,144] @ W_pre + b_pre  via fp32 WMMA, then scatter
//   sum/sumsq (atomicAdd) and max/min (f32 atomic max/min) into per-node stats.
// ---------------------------------------------------------------------------
__global__ __launch_bounds__(128) void pna_edge_kernel(
    const float* __restrict__ h, const float* __restrict__ ef,
    const long long* __restrict__ src, const long long* __restrict__ dst,
    const float* __restrict__ bpre, const float* __restrict__ swpre,
    float* __restrict__ sum, float* __restrict__ sumsq,
    float* __restrict__ mx, float* __restrict__ mn, float* __restrict__ deg) {
  __shared__ float zlds[4][16 * ZPITCH];
  __shared__ int   dlds[4][16];

  const int wave = threadIdx.x >> 5;
  const int lane = threadIdx.x & 31;
  const long long ebase = (long long)blockIdx.x * 64 + (long long)wave * 16;

  float* zw = zlds[wave];
  if (lane < 16) {
    long long e = ebase + lane;
    if (e >= N_EDGES) e = N_EDGES - 1;
    dlds[wave][lane] = (int)dst[e];
  }
  {
    int cr = -1;
    long long cs = 0, cd = 0, ce = 0;
    for (int i = lane; i < 16 * K_PRE; i += 32) {
      int r = i / K_PRE;
      int c = i - r * K_PRE;
      if (r != cr) {  // row index changes every ~4-5 iterations; cache src/dst
        cr = r;
        ce = ebase + r;
        if (ce >= N_EDGES) ce = N_EDGES - 1;
        cs = src[ce];
        cd = dst[ce];
      }
      float v;
      if (c < F_IN)          v = h[cs * F_IN + c];
      else if (c < 2 * F_IN) v = h[cd * F_IN + (c - F_IN)];
      else                   v = ef[ce * F_ED + (c - 2 * F_IN)];
      zw[r * ZPITCH + c] = v;
    }
  }
  __syncthreads();

  v8f acc[4] = {};
  const int mrow = lane & 15;
  const int koff = (lane >= 16) ? 2 : 0;
  const float* zrow = &zw[mrow * ZPITCH + koff];
  const int boff = mrow * 4 + koff;

  for (int kg = 0; kg < KG_PRE; ++kg) {
    v2f a = *(const v2f*)&zrow[kg * 4];
    const float* bp = &swpre[kg * 256 + boff];
#pragma unroll
    for (int nt = 0; nt < 4; ++nt)
      acc[nt] = wmma_f32(a, *(const v2f*)&bp[nt * 64], acc[nt]);
  }

  // Scatter: D layout -> VGPR r: lanes0-15 row M=r, lanes16-31 row M=r+8; col = nt*16 + lane%16
#pragma unroll
  for (int nt = 0; nt < 4; ++nt) {
    const int n = nt * 16 + mrow;
    const float bb = bpre[n];
#pragma unroll
    for (int r = 0; r < 8; ++r) {
      const int m = r + ((lane >= 16) ? 8 : 0);
      const long long e = ebase + m;
      const float v = acc[nt][r] + bb;
      if (e < N_EDGES) {
        const long long o = (long long)dlds[wave][m] * F_IN + n;
        atomicAdd(&sum[o], v);
        atomicAdd(&sumsq[o], v * v);
        unsafeAtomicMax(&mx[o], v);
        unsafeAtomicMin(&mn[o], v);
      }
    }
  }
  if (lane < 16) {
    long long e = ebase + lane;
    if (e < N_EDGES) atomicAdd(&deg[dlds[wave][lane]], 1.0f);
  }
}

// ---------------------------------------------------------------------------
// node kernel: 2 waves/block, 16 nodes per wave.
//   Stage h (16x64) and agg=[mean,max,min,std] (16x256) in LDS; amp/att are
//   realized by scaling the A operand with per-row scalars (lane's row is
//   fixed = lane%16), so the K=832 GEMM needs only K=320 of staging.
// ---------------------------------------------------------------------------
__global__ __launch_bounds__(64) void pna_node_kernel(
    const float* __restrict__ h, const float* __restrict__ snorm,
    const float* __restrict__ bpost, const float* __restrict__ swpost,
    const float* __restrict__ sum, const float* __restrict__ sumsq,
    const float* __restrict__ mx, const float* __restrict__ mn,
    const float* __restrict__ deg, float* __restrict__ out) {
  __shared__ float shh[2][16 * HPITCH];
  __shared__ float sha[2][16 * APITCH];

  const int wave = threadIdx.x >> 5;
  const int lane = threadIdx.x & 31;
  const int nbase = blockIdx.x * 32 + wave * 16;

  float* hw = shh[wave];
  float* aw = sha[wave];

  for (int i = lane; i < 16 * F_IN; i += 32) {
    int m = i >> 6, c = i & 63;
    int node = nbase + m;
    if (node >= N_NODES) node = N_NODES - 1;
    hw[m * HPITCH + c] = h[(long long)node * F_IN + c];
  }
  for (int i = lane; i < 16 * F_IN; i += 32) {
    int m = i >> 6, c = i & 63;
    int node = nbase + m;
    if (node >= N_NODES) node = N_NODES - 1;
    const long long o = (long long)node * F_IN + c;
    const float dg = deg[node];
    const float dc = fmaxf(dg, 1.0f);
    const float mean = sum[o] / dc;
    const float msq  = sumsq[o] / dc;
    const float var  = fmaxf(msq - mean * mean, 0.0f);
    const float sd   = sqrtf(var + PNA_EPS);
    const float has  = (dg > 0.0f) ? 1.0f : 0.0f;
    aw[m * APITCH + c]       = mean  * has;
    aw[m * APITCH + 64 + c]  = mx[o] * has;
    aw[m * APITCH + 128 + c] = mn[o] * has;
    aw[m * APITCH + 192 + c] = sd    * has;
  }

  const int mrow = lane & 15;
  int nodem = nbase + mrow;
  if (nodem >= N_NODES) nodem = N_NODES - 1;
  const float dcm  = fmaxf(deg[nodem], 1.0f);
  const float logd = logf(dcm + 1.0f);
  const float a_s  = logd / AVG_D_LOG;   // amplification scaler for this row
  const float t_s  = AVG_D_LOG / logd;   // attenuation scaler for this row
  __syncthreads();

  v8f acc[4] = {};
  const int koff = (lane >= 16) ? 2 : 0;
  const int boff = mrow * 4 + koff;
  const float* hr = &hw[mrow * HPITCH + koff];
  const float* ar = &aw[mrow * APITCH + koff];

  // h block: kgroups 0..15 of W_post
  for (int kg = 0; kg < 16; ++kg) {
    v2f a = *(const v2f*)&hr[kg * 4];
    const float* bp = &swpost[kg * 256 + boff];
#pragma unroll
    for (int nt = 0; nt < 4; ++nt)
      acc[nt] = wmma_f32(a, *(const v2f*)&bp[nt * 64], acc[nt]);
  }
  // agg/amp/att blocks: kgroups 16..79 / 80..143 / 144..207, one A load each
  for (int kg = 0; kg < 64; ++kg) {
    v2f a = *(const v2f*)&ar[kg * 4];
    v2f aamp = a * a_s;
    v2f aatt = a * t_s;
    const float* bg = &swpost[(16  + kg) * 256 + boff];
    const float* bm = &swpost[(80  + kg) * 256 + boff];
    const float* bt = &swpost[(144 + kg) * 256 + boff];
#pragma unroll
    for (int nt = 0; nt < 4; ++nt)
      acc[nt] = wmma_f32(a, *(const v2f*)&bg[nt * 64], acc[nt]);
#pragma unroll
    for (int nt = 0; nt < 4; ++nt)
      acc[nt] = wmma_f32(aamp, *(const v2f*)&bm[nt * 64], acc[nt]);
#pragma unroll
    for (int nt = 0; nt < 4; ++nt)
      acc[nt] = wmma_f32(aatt, *(const v2f*)&bt[nt * 64], acc[nt]);
  }

#pragma unroll
  for (int nt = 0; nt < 4; ++nt) {
    const int n = nt * 16 + mrow;
    const float bb = bpost[n];
#pragma unroll
    for (int r = 0; r < 8; ++r) {
      const int m = r + ((lane >= 16) ? 8 : 0);
      const int node = nbase + m;
      if (node < N_NODES)
        out[(long long)node * F_OUT + n] = (acc[nt][r] + bb) * snorm[node];
    }
  }
}

// ---------------------------------------------------------------------------
extern "C" void kernel_launch(void* const* d_in, const int* in_sizes, int n_in,
                              void* d_out, int out_size, void* d_ws, size_t ws_size,
                              hipStream_t stream) {
  const float*     h      = (const float*)d_in[0];
  const float*     ef     = (const float*)d_in[1];
  const float*     snorm  = (const float*)d_in[2];
  const float*     Wpre   = (const float*)d_in[3];
  const float*     bpre   = (const float*)d_in[4];
  const float*     Wpost  = (const float*)d_in[5];
  const float*     bpost  = (const float*)d_in[6];
  const long long* src    = (const long long*)d_in[7];
  const long long* dst    = (const long long*)d_in[8];
  float*           out    = (float*)d_out;

  // Workspace layout (floats): 4 per-node stat planes + degree + swizzled weights
  float* ws     = (float*)d_ws;
  float* sum    = ws;
  float* sumsq  = sum    + (long long)N_NODES * F_IN;
  float* mx     = sumsq  + (long long)N_NODES * F_IN;
  float* mn     = mx     + (long long)N_NODES * F_IN;
  float* deg    = mn     + (long long)N_NODES * F_IN;
  float* swpre  = deg    + N_NODES;
  float* swpost = swpre  + KG_PRE * 256;

  pna_prep_kernel<<<4096, 256, 0, stream>>>(Wpre, Wpost, sum, sumsq, mx, mn, deg,
                                            swpre, swpost);

  const int edge_blocks = (N_EDGES + 63) / 64;   // 12500, 64 edges per block
  pna_edge_kernel<<<edge_blocks, 128, 0, stream>>>(h, ef, src, dst, bpre, swpre,
                                                   sum, sumsq, mx, mn, deg);

  const int node_blocks = (N_NODES + 31) / 32;   // 1563, 32 nodes per block
  pna_node_kernel<<<node_blocks, 64, 0, stream>>>(h, snorm, bpost, swpost,
                                                  sum, sumsq, mx, mn, deg, out);
}